// KnnPooling_84052509982723
// MI455X (gfx1250) — compile-verified
//
#include <hip/hip_runtime.h>
#include <stdint.h>

// KNN max-pooling for MI455X (gfx1250, wave32).
// Gather path: CDNA5 async global->LDS DMA (GLOBAL_LOAD_ASYNC_TO_LDS_B128,
// tracked by ASYNCcnt), then LDS max-reduce. No matmul in this op => WMMA
// does not apply; the async-tensor/async-LDS engine is the CDNA5 feature
// that matches an irregular-gather workload.

typedef __attribute__((ext_vector_type(4))) float v4f;

#define NS 16                 // neighbors per query
#define CH 64                 // channels
#define WAVES_PER_BLOCK 4
#define THREADS (WAVES_PER_BLOCK * 32)

__global__ __launch_bounds__(THREADS)
void knn_max_pool_kernel(const float* __restrict__ feat,
                         const int* __restrict__ idx,
                         float* __restrict__ out,
                         int M) {
  // Per-wave staging buffer: 2 queries x 16 neighbor rows x 256B = 8KB/wave.
  __shared__ float smem[WAVES_PER_BLOCK * 2 * NS * CH];

  const int lane = threadIdx.x & 31;
  const int wv   = threadIdx.x >> 5;
  const int half = lane >> 4;    // which query of the pair this lane serves
  const int l15  = lane & 15;    // position within the 16-lane half

  float* wbuf = smem + wv * (2 * NS * CH);

  // LDS byte address of this lane's 16B chunk base for its query half.
  // (low 32 bits of a generic pointer to LDS == wave-relative LDS address)
  const uint32_t lds_base =
      (uint32_t)(uintptr_t)(wbuf + half * NS * CH) + (uint32_t)(l15 * 16);
  const uint64_t g_base =
      (uint64_t)(uintptr_t)feat + (uint64_t)(l15 * 16);

  const int pairs  = (M + 1) >> 1;
  const int nWaves = gridDim.x * WAVES_PER_BLOCK;
  const int wave0  = blockIdx.x * WAVES_PER_BLOCK + wv;

  for (int p = wave0; p < pairs; p += nWaves) {
    const int  q   = 2 * p + half;
    const bool qok = (q < M);

    // Lane l15 of each half holds neighbor l15's point index for its query.
    int myIdx = 0;
    if (qok) myIdx = idx[q * NS + l15];

    // Issue 16 async DMA gathers. Op k copies feature row feat[idx[q][k]]
    // (256B) into LDS: 16 consecutive lanes x 16B per query half => fully
    // coalesced 4x128B-line fetch; two queries serviced per instruction.
#pragma unroll
    for (int k = 0; k < NS; ++k) {
      const int      rk = __shfl(myIdx, (half << 4) + k, 32);
      const uint64_t ga = g_base + ((uint64_t)(uint32_t)rk << 8); // *256B
      const uint32_t la = lds_base + (uint32_t)(k * (CH * 4));
      asm volatile("global_load_async_to_lds_b128 %0, %1, off"
                   :: "v"(la), "v"(ga)
                   : "memory");
    }

    // Wait for this wave's async copies to land in LDS (ASYNCcnt -> 0).
    asm volatile("s_wait_asynccnt 0" ::: "memory");

    // Max-reduce the 16 neighbor rows out of LDS; each lane owns 4 channels.
    const float* buf = wbuf + half * NS * CH + l15 * 4;
    v4f acc = *(const v4f*)(buf);
#pragma unroll
    for (int k = 1; k < NS; ++k) {
      v4f v = *(const v4f*)(buf + k * CH);
      acc.x = fmaxf(acc.x, v.x);
      acc.y = fmaxf(acc.y, v.y);
      acc.z = fmaxf(acc.z, v.z);
      acc.w = fmaxf(acc.w, v.w);
    }

    if (qok) {
      // Output has zero reuse: non-temporal store keeps L2 for `feat`.
      __builtin_nontemporal_store(acc, (v4f*)(out + q * CH + l15 * 4));
    }
  }
}

extern "C" void kernel_launch(void* const* d_in, const int* in_sizes, int n_in,
                              void* d_out, int out_size, void* d_ws, size_t ws_size,
                              hipStream_t stream) {
  const float* feat = (const float*)d_in[0];   // [N_POINTS, 64] f32
  const int*   idx  = (const int*)d_in[1];     // [M, 16] int32 (JAX x64 off)
  float*       out  = (float*)d_out;           // [M, 64] f32

  const int M     = in_sizes[1] / NS;
  const int pairs = (M + 1) / 2;

  // ~4 query-pairs per wave of work, grid-stride loop handles remainder.
  int wavesWanted = (pairs + 3) / 4;
  int blocks = (wavesWanted + WAVES_PER_BLOCK - 1) / WAVES_PER_BLOCK;
  if (blocks < 1) blocks = 1;

  knn_max_pool_kernel<<<blocks, THREADS, 0, stream>>>(feat, idx, out, M);
}